// InteractionModule_5789615915708
// MI455X (gfx1250) — compile-verified
//
#include <hip/hip_runtime.h>
#include <hip/hip_bf16.h>

// ---------------------------------------------------------------------------
// PhysNet InteractionModule on MI455X (gfx1250, wave32, WMMA).
// All GEMMs run on the fp32 matrix pipe via V_WMMA_F32_16X16X4_F32 so results
// stay numerically equivalent to the fp32 reference. F=128, K=64 hardcoded.
// ---------------------------------------------------------------------------

typedef __attribute__((ext_vector_type(2))) float v2f;
typedef __attribute__((ext_vector_type(8))) float v8f;

__device__ __forceinline__ v8f wmma_k4(v2f a, v2f b, v8f c) {
  // D(16x16,f32) = A(16x4,f32) * B(4x16,f32) + C
  return __builtin_amdgcn_wmma_f32_16x16x4_f32(
      /*neg_a=*/false, a, /*neg_b=*/false, b,
      /*c_mod=*/(short)0, c, /*reuse_a=*/false, /*reuse_b=*/false);
}

__device__ __forceinline__ float ssp(float x) {
  // shifted softplus: log(1+exp(x)) - log(2), numerically stable form
  float e = __expf(-fabsf(x));
  return fmaxf(x, 0.0f) + __logf(1.0f + e) - 0.69314718055994531f;
}

#define F 128
#define KRBF 64
#define A_STRIDE 132   // 132 % 64 == 4 -> conflict-free bank pattern (4m+c)
#define E_STRIDE 68    // 68  % 64 == 4 -> same trick for the 16x64 RBF tile

// ---------------------------------------------------------------------------
// Elementwise pre-activation: xa = ssp(x)
// ---------------------------------------------------------------------------
__global__ void ssp_kernel(const float* __restrict__ x, float* __restrict__ xa, int n) {
  int i = blockIdx.x * 256 + threadIdx.x;
  if (i < n) xa[i] = ssp(x[i]);
}

// ---------------------------------------------------------------------------
// Edge message + scatter:
//   msg = ssp(xa[src] @ W_diff^T + b_diff) * (edge_attr @ W_G^T)
//   aggr[dst] += msg      (global f32 atomics)
// One block = 16 edges; 8 waves each own a 16x16 output tile (128 cols).
// ---------------------------------------------------------------------------
__global__ void __launch_bounds__(256)
edge_msg_kernel(const float* __restrict__ xa, const int* __restrict__ eidx,
                const float* __restrict__ ea, const float* __restrict__ Wd,
                const float* __restrict__ bd, const float* __restrict__ Wg,
                float* __restrict__ aggr, int E) {
  __shared__ __align__(16) float As[16 * A_STRIDE];   // gathered xa rows
  __shared__ __align__(16) float Es[16 * E_STRIDE];   // edge_attr rows
  __shared__ int dsts[16];

  const int e0 = blockIdx.x * 16;

  if (threadIdx.x < 16) {
    int e = e0 + threadIdx.x;
    if (e >= E) e = E - 1;
    dsts[threadIdx.x] = eidx[E + e];            // row 1 of edge_index = dst
  }
  // Gather 16 source-node feature rows (512B each; node table lives in L2).
  for (int i = threadIdx.x; i < 16 * F; i += 256) {
    int r = i >> 7, c = i & (F - 1);
    int e = e0 + r;
    if (e >= E) e = E - 1;
    int s = eidx[e];                            // row 0 of edge_index = src
    As[r * A_STRIDE + c] = xa[s * F + c];
  }
  for (int i = threadIdx.x; i < 16 * KRBF; i += 256) {
    int r = i >> 6, c = i & (KRBF - 1);
    int e = e0 + r;
    if (e >= E) e = E - 1;
    Es[r * E_STRIDE + c] = ea[e * KRBF + c];
  }
  __syncthreads();

  const int lane = threadIdx.x & 31;
  const int wv   = threadIdx.x >> 5;
  const int m    = lane & 15;       // A row within tile / D column index
  const int hi   = lane >> 4;       // half-wave select
  const int n    = wv * 16 + m;     // output feature column [0,128)

  // H1 = xa[src] @ W_diff^T : K = 128 -> 32 k-steps of 4
  v8f acc = {};
  const float* wrow = Wd + n * F;   // B[k][n] = W_diff[n][k]
#pragma unroll
  for (int kk = 0; kk < F / 4; ++kk) {
    v2f a = *(const v2f*)(&As[m * A_STRIDE + kk * 4 + 2 * hi]);
    v2f b = *(const v2f*)(&wrow[kk * 4 + 2 * hi]);
    acc = wmma_k4(a, b, acc);
  }

  // G = edge_attr @ W_G^T : K = 64 -> 16 k-steps of 4
  v8f acc2 = {};
  const float* grow = Wg + n * KRBF;
#pragma unroll
  for (int kk = 0; kk < KRBF / 4; ++kk) {
    v2f a = *(const v2f*)(&Es[m * E_STRIDE + kk * 4 + 2 * hi]);
    v2f b = *(const v2f*)(&grow[kk * 4 + 2 * hi]);
    acc2 = wmma_k4(a, b, acc2);
  }

  const float bn = bd[n];
#pragma unroll
  for (int j = 0; j < 8; ++j) {
    int r = j + 8 * hi;                          // edge row within tile
    int e = e0 + r;
    float msg = ssp(acc[j] + bn) * acc2[j];
    if (e < E) atomicAdd(&aggr[dsts[r] * F + n], msg);
  }
}

// ---------------------------------------------------------------------------
// Generic node GEMM: G = act_in(In) @ W^T ; per-tile post-op:
//   MODE 0: Out = ssp(G+b) + Ex1[row]        (update: + aggregated messages)
//   MODE 1: Out = G + b                      (residual hidden)
//   MODE 2: Out = Ex1[row] + G + b           (residual accumulate)
//   MODE 3: Out = G + b + Ex1[row]*Ex2[col]  (final: + x*u)
// WRITE2: also mirror result into Out2 (used to seed residual chain).
// ---------------------------------------------------------------------------
template <bool IN_ACT, int MODE, bool WRITE2>
__global__ void __launch_bounds__(256)
gemm128_kernel(const float* __restrict__ In, const float* __restrict__ W,
               const float* __restrict__ bias, const float* __restrict__ Ex1,
               const float* __restrict__ Ex2, float* __restrict__ Out,
               float* __restrict__ Out2, int N) {
  __shared__ __align__(16) float As[16 * A_STRIDE];

  const int row0 = blockIdx.x * 16;
  for (int i = threadIdx.x; i < 16 * F; i += 256) {
    int r = i >> 7, c = i & (F - 1);
    int rr = row0 + r;
    if (rr >= N) rr = N - 1;
    float v = In[rr * F + c];
    if (IN_ACT) v = ssp(v);
    As[r * A_STRIDE + c] = v;
  }
  __syncthreads();

  const int lane = threadIdx.x & 31;
  const int wv   = threadIdx.x >> 5;
  const int m    = lane & 15;
  const int hi   = lane >> 4;
  const int n    = wv * 16 + m;

  v8f acc = {};
  const float* wrow = W + n * F;
#pragma unroll
  for (int kk = 0; kk < F / 4; ++kk) {
    v2f a = *(const v2f*)(&As[m * A_STRIDE + kk * 4 + 2 * hi]);
    v2f b = *(const v2f*)(&wrow[kk * 4 + 2 * hi]);
    acc = wmma_k4(a, b, acc);
  }

  const float bn = bias[n];
#pragma unroll
  for (int j = 0; j < 8; ++j) {
    int r = row0 + j + 8 * hi;
    if (r >= N) continue;                        // uniform-enough tail guard
    float g = acc[j] + bn;
    float o;
    if (MODE == 0)      o = ssp(g) + Ex1[r * F + n];
    else if (MODE == 1) o = g;
    else if (MODE == 2) o = Ex1[r * F + n] + g;
    else                o = g + Ex1[r * F + n] * Ex2[n];
    Out[r * F + n] = o;
    if (WRITE2) Out2[r * F + n] = o;
  }
}

// ---------------------------------------------------------------------------
extern "C" void kernel_launch(void* const* d_in, const int* in_sizes, int n_in,
                              void* d_out, int out_size, void* d_ws, size_t ws_size,
                              hipStream_t stream) {
  const float* x      = (const float*)d_in[0];
  const int*   eidx   = (const int*)  d_in[1];   // [2,E]: row0=src, row1=dst
  const float* ea     = (const float*)d_in[2];
  const float* W_same = (const float*)d_in[3];
  const float* b_same = (const float*)d_in[4];
  const float* W_diff = (const float*)d_in[5];
  const float* b_diff = (const float*)d_in[6];
  const float* W_G    = (const float*)d_in[7];
  const float* rW1    = (const float*)d_in[8];
  const float* rb1    = (const float*)d_in[9];
  const float* rW2    = (const float*)d_in[10];
  const float* rb2    = (const float*)d_in[11];
  const float* W_last = (const float*)d_in[12];
  const float* b_last = (const float*)d_in[13];
  const float* u      = (const float*)d_in[14];

  const int NF = in_sizes[0];            // N * 128
  const int N  = NF / F;                 // 50000
  const int E  = in_sizes[2] / KRBF;     // 600000
  const int R  = in_sizes[8] / (F * F);  // 3

  float* out_final = (float*)d_out;        // output 0: v + x*u
  float* out_msged = (float*)d_out + NF;   // output 1: msged_x

  float* buf0 = (float*)d_ws;              // xa, then residual state t
  float* buf1 = buf0 + NF;                 // aggr, then residual hidden h

  const int nbN = (N + 15) / 16;
  const int nbE = (E + 15) / 16;

  // aggr = 0
  hipMemsetAsync(buf1, 0, (size_t)NF * sizeof(float), stream);

  // xa = ssp(x)
  ssp_kernel<<<(NF + 255) / 256, 256, 0, stream>>>(x, buf0, NF);

  // aggr[dst] += ssp(xa[src] @ W_diff^T + b_diff) * (edge_attr @ W_G^T)
  edge_msg_kernel<<<nbE, 256, 0, stream>>>(buf0, eidx, ea, W_diff, b_diff, W_G,
                                           buf1, E);

  // msged_x = ssp(xa @ W_same^T + b_same) + aggr  -> out_msged, and t=buf0
  gemm128_kernel<false, 0, true><<<nbN, 256, 0, stream>>>(
      buf0, W_same, b_same, buf1, nullptr, out_msged, buf0, N);

  // residual blocks: t = t + (ssp(ssp(t) @ W1^T + b1) @ W2^T + b2)
  for (int r = 0; r < R; ++r) {
    gemm128_kernel<true, 1, false><<<nbN, 256, 0, stream>>>(
        buf0, rW1 + (size_t)r * F * F, rb1 + (size_t)r * F,
        nullptr, nullptr, buf1, nullptr, N);
    gemm128_kernel<true, 2, false><<<nbN, 256, 0, stream>>>(
        buf1, rW2 + (size_t)r * F * F, rb2 + (size_t)r * F,
        buf0, nullptr, buf0, nullptr, N);
  }

  // out = ssp(t) @ W_last^T + b_last + x*u
  gemm128_kernel<true, 3, false><<<nbN, 256, 0, stream>>>(
      buf0, W_last, b_last, x, u, out_final, nullptr, N);
}